// Attention_62569083568830
// MI455X (gfx1250) — compile-verified
//
#include <hip/hip_runtime.h>
#include <math.h>

#define Bz 32
#define Tz 12
#define Nz 325
#define Kp 328             // padded K for N-dim contractions (zero-filled)
#define Dz 64
#define Hz 4
#define HDz 16
#define MSP 32
#define MT 6
#define BT (Bz * Tz)       // 384
#define SCL 0.25f          // HD^-0.5
#define GRAM_BLOCKS 128

typedef float v2f __attribute__((ext_vector_type(2)));
typedef float v8f __attribute__((ext_vector_type(8)));

// ---------------------------------------------------------------------------
// CDNA5 WMMA f32 16x16x4 (wave32). D = A(16x4) * B(4x16) + C(16x16).
// A frag: lanes 0-15 hold row M=lane, {K=k0,k0+1}; lanes 16-31 {K=k0+2,k0+3}.
// B frag: lanes 0-15 hold col N=lane, {K=k0,k0+1}; lanes 16-31 {K=k0+2,k0+3}.
// C/D   : VGPR v holds M=v (lanes 0-15) / M=v+8 (lanes 16-31), N = lane&15.
// Bounds policy: A rows are address-clamped (garbage rows are never stored);
// K overruns hit zero-padded columns, so no masks and no branches in K loops.
// ---------------------------------------------------------------------------
__device__ __forceinline__ v8f wmma4(v2f a, v2f b, v8f c) {
  return __builtin_amdgcn_wmma_f32_16x16x4_f32(false, a, false, b, (short)0, c,
                                               false, false);
}

__device__ __forceinline__ void stC(float* C, int ldc, int row0, int col0,
                                    int rowMax, v8f acc, float scale,
                                    bool accum) {
  int lane = threadIdx.x & 31;
  int c = col0 + (lane & 15);
  int rb = row0 + ((lane >> 4) << 3);
#pragma unroll
  for (int v = 0; v < 8; ++v) {
    int r = rb + v;
    if (r < rowMax) {
      float val = acc[v] * scale;
      if (accum) C[r * ldc + c] += val;
      else C[r * ldc + c] = val;
    }
  }
}

// ------------------------------- prep kernels ------------------------------

// x -> xpad (adds a zeroed 256-float tail so K-overrun reads stay in-bounds)
__global__ void k_xpad(const float* __restrict__ x, float* xpad, int total) {
  int i = blockIdx.x * 256 + threadIdx.x;
  if (i < total) xpad[i] = x[i];
  else if (i < total + 256) xpad[i] = 0.0f;
}

// a[n,k] = adj[n,k] / sum_k adj[n,k], stored 325 x 328 with zero pad cols
__global__ void k_adj_norm(const float* __restrict__ adj, float* a_norm) {
  int row = blockIdx.x;
  __shared__ float red[256];
  float s = 0.0f;
  for (int c = threadIdx.x; c < Nz; c += 256) s += adj[row * Nz + c];
  red[threadIdx.x] = s;
  __syncthreads();
  for (int st = 128; st > 0; st >>= 1) {
    if ((int)threadIdx.x < st) red[threadIdx.x] += red[threadIdx.x + st];
    __syncthreads();
  }
  float inv = 1.0f / red[0];
  for (int c = threadIdx.x; c < Kp; c += 256)
    a_norm[row * Kp + c] = (c < Nz) ? adj[row * Nz + c] * inv : 0.0f;
}

// Wc[e,d] = sum_f W_mlp[e,f] * W_fc1[f,d]
__global__ void k_wc(const float* __restrict__ Wmlp,
                     const float* __restrict__ Wfc1, float* Wc) {
  for (int i = threadIdx.x; i < 4096; i += 256) {
    int e = i >> 6, d = i & 63;
    float s = 0.0f;
    for (int f = 0; f < 64; ++f) s += Wmlp[e * 64 + f] * Wfc1[f * 64 + d];
    Wc[i] = s;
  }
}

// Forward DFT rows (cos / -sin) at sp_modes, 64 x 328 zero-padded;
// irfft matrix IFsp 325 x 64.
__global__ void k_dftmats(const int* __restrict__ sp_modes, float* Fsp,
                          float* IFsp) {
  int i = blockIdx.x * 256 + threadIdx.x;
  const int total1 = 64 * Kp;
  const float twopi = 6.28318530717958647692f;
  if (i < total1) {
    int cm = i / Kp, n = i % Kp;
    int m = cm & 31;
    float v = 0.0f;
    if (n < Nz) {
      float th = twopi * (float)sp_modes[m] * (float)n / (float)Nz;
      v = (cm >= 32) ? -sinf(th) : cosf(th);
    }
    Fsp[i] = v;
  } else if (i < total1 + Nz * 64) {
    int j = i - total1;
    int n = j / 64, cm = j % 64;
    int m = cm & 31;
    int f = sp_modes[m];
    float w = (f == 0) ? 1.0f : 2.0f;  // N=325 odd -> no Nyquist bin
    float th = twopi * (float)f * (float)n / (float)Nz;
    float sv = (cm >= 32) ? -sinf(th) : cosf(th);
    IFsp[n * 64 + cm] = w * sv / (float)Nz;
  }
}

// Temporal branch collapses to vf/MT (softmax rows sum to 1):
// K_time[t,t'] = (1/(MT*T)) sum_m w_m cos(2pi f_m (t-t')/T)
__global__ void k_ktime(const int* __restrict__ t_modes, float* Kt) {
  int i = threadIdx.x;
  if (i < Tz * Tz) {
    int t = i / Tz, tp = i % Tz;
    float s = 0.0f;
    for (int m = 0; m < MT; ++m) {
      int f = t_modes[m];
      float w = (f == 0 || 2 * f == Tz) ? 1.0f : 2.0f;
      s += w * cosf(6.28318530717958647692f * (float)f * (float)(t - tp) /
                    (float)Tz);
    }
    Kt[i] = s / (float)(MT * Tz);
  }
}

__global__ void k_absw(const float* __restrict__ wQ, float* absWq) {
  int i = blockIdx.x * 256 + threadIdx.x;
  if (i < MSP * (MSP - 1) * HDz) absWq[i] = fabsf(wQ[i]);
}

// Per-block partial Gram matrices G = x^T x over row chunks (deterministic).
__global__ void k_gram(const float* __restrict__ x, float* Gpart) {
  __shared__ float Gl[4096];
  __shared__ float xs[4 * 64];
  for (int i = threadIdx.x; i < 4096; i += 256) Gl[i] = 0.0f;
  const int R = Bz * Tz * Nz;
  const int chunk = (R + GRAM_BLOCKS - 1) / GRAM_BLOCKS;
  int r0 = blockIdx.x * chunk;
  int r1 = min(r0 + chunk, R);
  __syncthreads();
  for (int r = r0; r < r1; r += 4) {
    int nr = min(4, r1 - r);
    for (int i = threadIdx.x; i < nr * 64; i += 256)
      xs[i] = x[(size_t)r * 64 + i];
    __syncthreads();
    for (int i = threadIdx.x; i < 4096; i += 256) {
      int a = i >> 6, b = i & 63;
      float s = Gl[i];
      for (int rr = 0; rr < nr; ++rr) s += xs[rr * 64 + a] * xs[rr * 64 + b];
      Gl[i] = s;
    }
    __syncthreads();
  }
  for (int i = threadIdx.x; i < 4096; i += 256)
    Gpart[(size_t)blockIdx.x * 4096 + i] = Gl[i];
}

// ||x W^T||_F^2 = sum_e (W G W^T)_ee ; scal[0]=1/||q||, scal[1]=1/||k||
__global__ void k_norms(const float* __restrict__ Gpart,
                        const float* __restrict__ Wq,
                        const float* __restrict__ Wk, float* scal) {
  __shared__ float Gs[4096];
  __shared__ float red[256];
  for (int i = threadIdx.x; i < 4096; i += 256) {
    float s = 0.0f;
    for (int p = 0; p < GRAM_BLOCKS; ++p) s += Gpart[(size_t)p * 4096 + i];
    Gs[i] = s;
  }
  __syncthreads();
  float sq = 0.0f, sk = 0.0f;
  if (threadIdx.x < 64) {
    int e = threadIdx.x;
    for (int dp = 0; dp < 64; ++dp) {
      float tq = 0.0f, tk = 0.0f;
      for (int d = 0; d < 64; ++d) {
        float g = Gs[d * 64 + dp];
        tq += Wq[e * 64 + d] * g;
        tk += Wk[e * 64 + d] * g;
      }
      sq += Wq[e * 64 + dp] * tq;
      sk += Wk[e * 64 + dp] * tk;
    }
  }
  red[threadIdx.x] = sq;
  __syncthreads();
  for (int st = 128; st > 0; st >>= 1) {
    if ((int)threadIdx.x < st) red[threadIdx.x] += red[threadIdx.x + st];
    __syncthreads();
  }
  if (threadIdx.x == 0) scal[0] = rsqrtf(fmaxf(red[0], 1e-30f));
  __syncthreads();
  red[threadIdx.x] = sk;
  __syncthreads();
  for (int st = 128; st > 0; st >>= 1) {
    if ((int)threadIdx.x < st) red[threadIdx.x] += red[threadIdx.x + st];
    __syncthreads();
  }
  if (threadIdx.x == 0) scal[1] = rsqrtf(fmaxf(red[0], 1e-30f));
}

// ------------------------------- GEMM kernels ------------------------------

// out = (a_norm @ x_bt) @ Wc^T + b_mlp    (writes out: runs first)
__global__ void k_gcn(const float* __restrict__ a_norm,
                      const float* __restrict__ xpad,
                      const float* __restrict__ Wc,
                      const float* __restrict__ bml, float* out) {
  int bt = blockIdx.x, rt = blockIdx.y;
  int wave = threadIdx.x >> 5;
  int lane = threadIdx.x & 31;
  int la = lane & 15, kd = (lane >> 4) << 1;
  int row0 = rt * 16, col0 = wave * 16;
  const float* xb = xpad + (size_t)bt * Nz * Dz;

  // stage 1: agg tile = a_norm[row0:+16, :] @ x_bt[:, col0:+16]
  const float* Ap = a_norm + (size_t)min(row0 + la, Nz - 1) * Kp + kd;
  const float* Bp = xb + col0 + la + (size_t)kd * Dz;
  v8f acc = {};
#pragma unroll 4
  for (int k = 0; k < Kp; k += 4) {
    v2f a = *(const v2f*)(Ap + k);
    v2f b;
    b.x = Bp[(size_t)k * Dz];
    b.y = Bp[(size_t)(k + 1) * Dz];
    acc = wmma4(a, b, acc);
  }
  __shared__ alignas(16) float aggL[16 * 64];
  stC(aggL, 64, 0, col0, 16, acc, 1.0f, false);
  __syncthreads();

  // stage 2: out tile = aggL(16x64) @ Wc^T, + bias
  const float* Ap2 = aggL + la * 64 + kd;
  const float* Bp2 = Wc + (size_t)(col0 + la) * 64 + kd;  // W^T frag: contiguous
  v8f acc2 = {};
#pragma unroll
  for (int k = 0; k < 64; k += 4) {
    v2f a = *(const v2f*)(Ap2 + k);
    v2f b = *(const v2f*)(Bp2 + k);
    acc2 = wmma4(a, b, acc2);
  }
  int c = col0 + la;
  float bias = bml[c];
  int rb = row0 + ((lane >> 4) << 3);
  float* ob = out + (size_t)bt * Nz * Dz;
#pragma unroll
  for (int v = 0; v < 8; ++v) {
    int r = rb + v;
    if (r < Nz) ob[r * Dz + c] = acc2[v] + bias;
  }
}

// Fx[bt] = Fsp(64x325) @ x_bt(325x64)  -- shared DFT for q/k/v
__global__ void k_fx(const float* __restrict__ Fsp,
                     const float* __restrict__ xpad, float* Fx) {
  int bt = blockIdx.x;
  int tile = blockIdx.y * 8 + (threadIdx.x >> 5);
  int rt = tile >> 2, ct = tile & 3;
  int lane = threadIdx.x & 31;
  int la = lane & 15, kd = (lane >> 4) << 1;
  const float* xb = xpad + (size_t)bt * Nz * Dz;
  const float* Ap = Fsp + (size_t)(rt * 16 + la) * Kp + kd;
  const float* Bp = xb + ct * 16 + la + (size_t)kd * Dz;
  v8f acc = {};
#pragma unroll 4
  for (int k = 0; k < Kp; k += 4) {
    v2f a = *(const v2f*)(Ap + k);
    v2f b;
    b.x = Bp[(size_t)k * Dz];
    b.y = Bp[(size_t)(k + 1) * Dz];
    acc = wmma4(a, b, acc);
  }
  stC(Fx + (size_t)bt * 4096, 64, rt * 16, ct * 16, 64, acc, 1.0f, false);
}

// {Q,K,V}F[bt] = scale_tau * Fx[bt] @ W_tau^T   (scale = 1/||q||, 1/||k||, 1)
__global__ void k_qkvf(const float* __restrict__ Fx,
                       const float* __restrict__ Wq,
                       const float* __restrict__ Wk,
                       const float* __restrict__ Wv,
                       const float* __restrict__ scal, float* QF, float* KF,
                       float* VF) {
  int bt = blockIdx.x;
  int tau = blockIdx.y;
  const float* W = (tau == 0) ? Wq : (tau == 1) ? Wk : Wv;
  float sc = (tau == 0) ? scal[0] : (tau == 1) ? scal[1] : 1.0f;
  float* O = ((tau == 0) ? QF : (tau == 1) ? KF : VF) + (size_t)bt * 4096;
  const float* A = Fx + (size_t)bt * 4096;
  int lane = threadIdx.x & 31;
  int la = lane & 15, kd = (lane >> 4) << 1;
  int w = threadIdx.x >> 5;
  for (int t = w; t < 16; t += 8) {
    int rt = t >> 2, ct = t & 3;
    const float* Ap = A + (size_t)(rt * 16 + la) * 64 + kd;
    const float* Bp = W + (size_t)(ct * 16 + la) * 64 + kd;
    v8f acc = {};
#pragma unroll
    for (int k = 0; k < 64; k += 4) {
      v2f a = *(const v2f*)(Ap + k);
      v2f b = *(const v2f*)(Bp + k);
      acc = wmma4(a, b, acc);
    }
    stC(O, 64, rt * 16, ct * 16, 64, acc, sc, false);
  }
}

// Spatial attention core:
// OSP[j,hd] = vf[j,hd] * mean_m1 softmax_j(SCL*|kf_j|*|qcat_{m1,j}|)
__global__ void k_att_sp(const float* __restrict__ QF,
                         const float* __restrict__ KF,
                         const float* __restrict__ VF,
                         const float* __restrict__ absWq, float* OSP) {
  int g = blockIdx.x;
  int bt = g >> 2, h = g & 3;
  const float* QFb = QF + (size_t)bt * 4096;
  const float* KFb = KF + (size_t)bt * 4096;
  const float* VFb = VF + (size_t)bt * 4096;
  float* OSPb = OSP + (size_t)bt * 4096;
  __shared__ float kabs[MSP * HDz];
  __shared__ float qabs[MSP * HDz];
  __shared__ float rmax[MSP * HDz];
  __shared__ float rinv[MSP * HDz];
  int tid = threadIdx.x;
  for (int p = tid; p < MSP * HDz; p += 256) {
    int m = p >> 4, hd = p & 15;
    int col = h * HDz + hd;
    float kr = KFb[m * 64 + col], ki = KFb[(32 + m) * 64 + col];
    kabs[p] = SCL * sqrtf(kr * kr + ki * ki);
    float qr = QFb[m * 64 + col], qi = QFb[(32 + m) * 64 + col];
    qabs[p] = sqrtf(qr * qr + qi * qi);
  }
  __syncthreads();
  // phase 1: per (m1,hd): softmax stats over j in [0,32)
  for (int p = tid; p < MSP * HDz; p += 256) {
    int m1 = p >> 4, hd = p & 15;
    float qa = qabs[p];
    const float* wr = absWq + (size_t)(m1 * (MSP - 1)) * HDz + hd;
    float mx = kabs[hd] * qa;  // j = 0 uses |qf|
#pragma unroll
    for (int j = 1; j < MSP; ++j)
      mx = fmaxf(mx, kabs[j * HDz + hd] * wr[(j - 1) * HDz]);
    float s = __expf(kabs[hd] * qa - mx);
#pragma unroll
    for (int j = 1; j < MSP; ++j)
      s += __expf(kabs[j * HDz + hd] * wr[(j - 1) * HDz] - mx);
    rmax[p] = mx;
    rinv[p] = 1.0f / s;
  }
  __syncthreads();
  // phase 2: per (j,hd): deterministic mean over m1, gate vf
  for (int p = tid; p < MSP * HDz; p += 256) {
    int j = p >> 4, hd = p & 15;
    float ka = kabs[p];
    float mw = 0.0f;
#pragma unroll
    for (int m1 = 0; m1 < MSP; ++m1) {
      int rp = m1 * HDz + hd;
      float qv = (j == 0) ? qabs[rp]
                          : absWq[(size_t)(m1 * (MSP - 1) + (j - 1)) * HDz + hd];
      mw += __expf(ka * qv - rmax[rp]) * rinv[rp];
    }
    mw *= (1.0f / (float)MSP);
    int col = h * HDz + hd;
    OSPb[j * 64 + col] = VFb[j * 64 + col] * mw;
    OSPb[(32 + j) * 64 + col] = VFb[(32 + j) * 64 + col] * mw;
  }
}

// out += IFsp(325x64) @ OSP_bt(64x64)   (irfft of gated modes)
__global__ void k_irfft_sp(const float* __restrict__ IFsp,
                           const float* __restrict__ OSP, float* out) {
  int bt = blockIdx.x, rt = blockIdx.y;
  int wave = threadIdx.x >> 5;
  int lane = threadIdx.x & 31;
  int la = lane & 15, kd = (lane >> 4) << 1;
  const float* Ob = OSP + (size_t)bt * 4096;
  const float* Ap = IFsp + (size_t)min(rt * 16 + la, Nz - 1) * 64 + kd;
  const float* Bp = Ob + wave * 16 + la + kd * 64;
  v8f acc = {};
#pragma unroll
  for (int k = 0; k < 64; k += 4) {
    v2f a = *(const v2f*)(Ap + k);
    v2f b;
    b.x = Bp[k * 64];
    b.y = Bp[(k + 1) * 64];
    acc = wmma4(a, b, acc);
  }
  stC(out + (size_t)bt * Nz * Dz, Dz, rt * 16, wave * 16, Nz, acc, 1.0f, true);
}

// xK[b,t,n,d] = sum_t' Ktime[t,t'] x[b,t',n,d]
__global__ void k_xk(const float* __restrict__ x, const float* __restrict__ Kt,
                     float* xK) {
  __shared__ float Kts[Tz * Tz];
  if (threadIdx.x < Tz * Tz) Kts[threadIdx.x] = Kt[threadIdx.x];
  __syncthreads();
  int i = blockIdx.x * 256 + threadIdx.x;
  if (i >= Bz * Nz * Dz) return;
  int d = i % Dz;
  int rem = i / Dz;
  int n = rem % Nz;
  int b = rem / Nz;
  float vbuf[Tz];
#pragma unroll
  for (int t = 0; t < Tz; ++t)
    vbuf[t] = x[((size_t)(b * Tz + t) * Nz + n) * Dz + d];
#pragma unroll
  for (int t = 0; t < Tz; ++t) {
    float s = 0.0f;
#pragma unroll
    for (int tp = 0; tp < Tz; ++tp) s += Kts[t * Tz + tp] * vbuf[tp];
    xK[((size_t)(b * Tz + t) * Nz + n) * Dz + d] = s;
  }
}

// out += xK_bt @ Wv_t^T   (temporal branch)
__global__ void k_tv(const float* __restrict__ xK,
                     const float* __restrict__ Wvt, float* out) {
  int bt = blockIdx.x, rt = blockIdx.y;
  int wave = threadIdx.x >> 5;
  int lane = threadIdx.x & 31;
  int la = lane & 15, kd = (lane >> 4) << 1;
  const float* A = xK + (size_t)bt * Nz * Dz;
  const float* Ap = A + (size_t)min(rt * 16 + la, Nz - 1) * 64 + kd;
  const float* Bp = Wvt + (size_t)(wave * 16 + la) * 64 + kd;
  v8f acc = {};
#pragma unroll
  for (int k = 0; k < 64; k += 4) {
    v2f a = *(const v2f*)(Ap + k);
    v2f b = *(const v2f*)(Bp + k);
    acc = wmma4(a, b, acc);
  }
  stC(out + (size_t)bt * Nz * Dz, Dz, rt * 16, wave * 16, Nz, acc, 1.0f, true);
}

// --------------------------------- launch ----------------------------------

extern "C" void kernel_launch(void* const* d_in, const int* in_sizes, int n_in,
                              void* d_out, int out_size, void* d_ws,
                              size_t ws_size, hipStream_t stream) {
  (void)in_sizes; (void)n_in; (void)out_size; (void)ws_size;
  const float* x       = (const float*)d_in[0];
  const float* adj     = (const float*)d_in[1];
  const float* Wq_geo  = (const float*)d_in[2];
  const float* Wk_geo  = (const float*)d_in[3];
  const float* Wv_geo  = (const float*)d_in[4];
  // d_in[5]=Wq_t, d_in[6]=Wk_t unused: temporal softmax rows sum to 1, so the
  // temporal attention output is exactly vf/M_T independent of q/k.
  const float* Wv_t    = (const float*)d_in[7];
  const float* W_fc1   = (const float*)d_in[8];
  const float* W_mlp   = (const float*)d_in[9];
  const float* b_mlp   = (const float*)d_in[10];
  const float* wQ      = (const float*)d_in[11];
  // d_in[12]=weights_Q_t unused for the same reason.
  const int* sp_modes  = (const int*)d_in[13];
  const int* t_modes   = (const int*)d_in[14];
  float* out = (float*)d_out;

  float* w = (float*)d_ws;
  size_t off = 0;
  auto alloc = [&](size_t nflt) {
    float* p = w + off;
    off += (nflt + 63) & ~(size_t)63;
    return p;
  };
  const int XTOT = BT * Nz * Dz;  // 7,987,200
  float* xpad   = alloc((size_t)XTOT + 256);
  float* a_norm = alloc((size_t)Nz * Kp);
  float* Wc     = alloc(4096);
  float* Fsp    = alloc((size_t)64 * Kp);
  float* IFsp   = alloc((size_t)Nz * 64);
  float* Kt     = alloc(Tz * Tz);
  float* absWq  = alloc((size_t)MSP * (MSP - 1) * HDz);
  float* Gpart  = alloc((size_t)GRAM_BLOCKS * 4096);
  float* scal   = alloc(64);
  float* Fx     = alloc((size_t)BT * 4096);
  float* QF     = alloc((size_t)BT * 4096);
  float* KF     = alloc((size_t)BT * 4096);
  float* VF     = alloc((size_t)BT * 4096);
  float* OSP    = alloc((size_t)BT * 4096);
  float* xK     = alloc((size_t)XTOT);

  // prep
  k_xpad<<<(XTOT + 256 + 255) / 256, 256, 0, stream>>>(x, xpad, XTOT);
  k_adj_norm<<<Nz, 256, 0, stream>>>(adj, a_norm);
  k_wc<<<1, 256, 0, stream>>>(W_mlp, W_fc1, Wc);
  k_dftmats<<<(64 * Kp + Nz * 64 + 255) / 256, 256, 0, stream>>>(sp_modes, Fsp,
                                                                 IFsp);
  k_ktime<<<1, 256, 0, stream>>>(t_modes, Kt);
  k_absw<<<(MSP * (MSP - 1) * HDz + 255) / 256, 256, 0, stream>>>(wQ, absWq);
  k_gram<<<GRAM_BLOCKS, 256, 0, stream>>>(x, Gpart);
  k_norms<<<1, 256, 0, stream>>>(Gpart, Wq_geo, Wk_geo, scal);

  // GCN writes out first
  k_gcn<<<dim3(BT, 21), 128, 0, stream>>>(a_norm, xpad, Wc, b_mlp, out);

  // spatial branch (accumulates into out)
  k_fx<<<dim3(BT, 2), 256, 0, stream>>>(Fsp, xpad, Fx);
  k_qkvf<<<dim3(BT, 3), 256, 0, stream>>>(Fx, Wq_geo, Wk_geo, Wv_geo, scal, QF,
                                          KF, VF);
  k_att_sp<<<BT * Hz, 256, 0, stream>>>(QF, KF, VF, absWq, OSP);
  k_irfft_sp<<<dim3(BT, 21), 128, 0, stream>>>(IFsp, OSP, out);

  // temporal branch (accumulates into out)
  k_xk<<<(Bz * Nz * Dz + 255) / 256, 256, 0, stream>>>(x, Kt, xK);
  k_tv<<<dim3(BT, 21), 128, 0, stream>>>(xK, Wv_t, out);
}